// GCL_18880676233903
// MI455X (gfx1250) — compile-verified
//
#include <hip/hip_runtime.h>
#include <cstdint>
#include <cstddef>

typedef __attribute__((ext_vector_type(2))) float v2f;
typedef __attribute__((ext_vector_type(8))) float v8f;

#define DIN 128
#define DOUT 128
#define BN_EPS 1e-5f

// ---------------------------------------------------------------------------
// Kernel 1: zero the x_sum accumulator (lives in d_ws; harness poisons it).
// ---------------------------------------------------------------------------
__global__ void gcl_zero_kernel(float4* __restrict__ p, int n4) {
  int i = blockIdx.x * blockDim.x + threadIdx.x;
  int stride = gridDim.x * blockDim.x;
  float4 z = {0.f, 0.f, 0.f, 0.f};
  for (; i < n4; i += stride) p[i] = z;
}

// ---------------------------------------------------------------------------
// Kernel 2: edge scatter.  One wave per edge; each lane handles 4 consecutive
// columns (32 lanes x 4 = 128).  x and x_sum are L2-resident (51.2 MB each vs
// 192 MB L2), so gathers and the native f32 atomics both run at L2 speed.
// ---------------------------------------------------------------------------
__global__ void gcl_scatter_kernel(const float4* __restrict__ x4,
                                   const float* __restrict__ eval,
                                   const int* __restrict__ erow,
                                   const int* __restrict__ ecol,
                                   float* __restrict__ xsum,
                                   int E) {
  int edge = (blockIdx.x * blockDim.x + threadIdx.x) >> 5;
  int lane = threadIdx.x & 31;
  if (edge >= E) return;
  int r = erow[edge];
  int c = ecol[edge];
  float v = eval[edge];
  float4 m = x4[(size_t)c * (DIN / 4) + lane];
  m.x *= v; m.y *= v; m.z *= v; m.w *= v;
  float* dst = xsum + (size_t)r * DIN + lane * 4;
  // Native L2 f32 atomics (no return -> STOREcnt; s_endpgm waits for idle).
  asm volatile(
      "global_atomic_add_f32 %0, %1, off\n\t"
      "global_atomic_add_f32 %0, %2, off offset:4\n\t"
      "global_atomic_add_f32 %0, %3, off offset:8\n\t"
      "global_atomic_add_f32 %0, %4, off offset:12"
      :
      : "v"(dst), "v"(m.x), "v"(m.y), "v"(m.z), "v"(m.w)
      : "memory");
}

// ---------------------------------------------------------------------------
// Kernel 3: fused GEMM (WMMA f32 16x16x4) + bias + BatchNorm(eval) + ReLU.
// Block = 256 threads = 8 waves.  Wave w owns columns [16w, 16w+16).
// Each wave preloads all 32 B-fragments of W (64 VGPRs) once and reuses them
// across TILES_PER_BLOCK row tiles; the 16x128 A tile is staged through LDS.
//
// WMMA f32 16x16x4 layouts (ISA 7.12.2):
//   A: lane l -> M = l&15, VGPR j holds K = 2*(l>>4)+j
//   B: lane l -> N = l&15, VGPR j holds K = 2*(l>>4)+j
//   C/D: lane l, VGPR r -> M = r + 8*(l>>4), N = l&15
// ---------------------------------------------------------------------------
__global__ void __launch_bounds__(256)
gcl_gemm_bn_relu_kernel(const float* __restrict__ xsum,
                        const float* __restrict__ Wm,
                        const float* __restrict__ bias,
                        const float* __restrict__ gamma,
                        const float* __restrict__ beta,
                        const float* __restrict__ rmean,
                        const float* __restrict__ rvar,
                        float* __restrict__ out,
                        int numRowTiles, int tilesPerBlock) {
  __shared__ float4 ldsA4[16 * DIN / 4];
  float* ldsA = (float*)ldsA4;

  const int wv = threadIdx.x >> 5;
  const int lane = threadIdx.x & 31;
  const int half = lane >> 4;   // 0 or 1
  const int mn = lane & 15;     // M (for A/C) or N (for B/C)
  const int col0 = wv * 16;
  const int n = col0 + mn;

  // Preload B fragments of W (row-major DIN x DOUT): k = 4*kk + 2*half + j.
  v2f bfr[32];
#pragma unroll
  for (int kk = 0; kk < 32; ++kk) {
    int k = kk * 4 + half * 2;
    v2f t;
    t.x = Wm[(size_t)(k + 0) * DOUT + n];
    t.y = Wm[(size_t)(k + 1) * DOUT + n];
    bfr[kk] = t;
  }

  // Fused epilogue constants for this output column:
  //   out = max( acc*s + ((b - mean)*s + beta), 0 ),  s = gamma*rsqrt(var+eps)
  const float s = gamma[n] * rsqrtf(rvar[n] + BN_EPS);
  const float tshift = (bias[n] - rmean[n]) * s + beta[n];

  for (int t = 0; t < tilesPerBlock; ++t) {
    int tile = blockIdx.x * tilesPerBlock + t;
    if (tile >= numRowTiles) break;
    int row0 = tile * 16;

    // Cooperative stage of the 16x128 f32 A tile (512 float4, 2 per thread).
    const float4* src = (const float4*)(xsum + (size_t)row0 * DIN);
    ldsA4[threadIdx.x * 2 + 0] = src[threadIdx.x * 2 + 0];
    ldsA4[threadIdx.x * 2 + 1] = src[threadIdx.x * 2 + 1];
    __syncthreads();

    v8f acc = {};
#pragma unroll
    for (int kk = 0; kk < 32; ++kk) {
      v2f a = *(const v2f*)&ldsA[mn * DIN + kk * 4 + half * 2];  // ds_load_b64
      acc = __builtin_amdgcn_wmma_f32_16x16x4_f32(
          /*neg_a=*/false, a, /*neg_b=*/false, bfr[kk],
          /*c_mod=*/(short)0, acc, /*reuse_a=*/false, /*reuse_b=*/false);
    }

#pragma unroll
    for (int r = 0; r < 8; ++r) {
      int m = row0 + r + half * 8;
      float y = acc[r] * s + tshift;
      out[(size_t)m * DOUT + n] = fmaxf(y, 0.f);
    }
    __syncthreads();  // protect LDS before next tile's stage
  }
}

// ---------------------------------------------------------------------------
// Host launcher.  Inputs (setup_inputs order):
//   0:x [N*128 f32] 1:edge_val [E f32] 2:W [128*128 f32] 3:b 4:gamma 5:beta
//   6:running_mean 7:running_var 8:edge_row [E i32] 9:edge_col [E i32]
// ---------------------------------------------------------------------------
extern "C" void kernel_launch(void* const* d_in, const int* in_sizes, int n_in,
                              void* d_out, int out_size, void* d_ws, size_t ws_size,
                              hipStream_t stream) {
  (void)n_in; (void)out_size; (void)ws_size;
  const float* x = (const float*)d_in[0];
  const float* edge_val = (const float*)d_in[1];
  const float* Wm = (const float*)d_in[2];
  const float* b = (const float*)d_in[3];
  const float* gamma = (const float*)d_in[4];
  const float* beta = (const float*)d_in[5];
  const float* rmean = (const float*)d_in[6];
  const float* rvar = (const float*)d_in[7];
  const int* erow = (const int*)d_in[8];
  const int* ecol = (const int*)d_in[9];

  const int N = in_sizes[0] / DIN;
  const int E = in_sizes[1];
  float* xsum = (float*)d_ws;  // N*DIN f32 accumulator (51.2 MB)

  // 1) zero accumulator
  int n4 = N * DIN / 4;
  gcl_zero_kernel<<<2048, 256, 0, stream>>>((float4*)xsum, n4);

  // 2) edge scatter: 8 edges (waves) per 256-thread block
  int gridS = (E + 7) / 8;
  gcl_scatter_kernel<<<gridS, 256, 0, stream>>>((const float4*)x, edge_val,
                                                erow, ecol, xsum, E);

  // 3) fused WMMA GEMM + BN + ReLU (N divisible by 16 for reference shape)
  int numRowTiles = N / 16;
  const int TPB = 5;
  int gridG = (numRowTiles + TPB - 1) / TPB;
  gcl_gemm_bn_relu_kernel<<<gridG, 256, 0, stream>>>(
      xsum, Wm, b, gamma, beta, rmean, rvar, (float*)d_out, numRowTiles, TPB);
}